// OldTransportMapKernel_6116033429508
// MI455X (gfx1250) — compile-verified
//
#include <hip/hip_runtime.h>
#include <math.h>

typedef __attribute__((ext_vector_type(2))) float v2f;
typedef __attribute__((ext_vector_type(4))) float v4f;
typedef __attribute__((ext_vector_type(8))) float v8f;

#define N_LOCS   4096
#define N_REPS   128
#define M_NBRS   48
#define XS_STRIDE 50   // 128x48 tile padded: 50*i mod 64 distinct over 16 lanes
#define S_STRIDE  129  // 129 mod 64 == 1 -> conflict-free column walks

// One workgroup (256 threads = 8 waves) per location.
// Phase 1: gather+scale Xs (128x48) into LDS.
// Phase 2: S = Xs * Xs^T via v_wmma_f32_16x16x4_f32 (wave w owns tile-row w).
// Phase 3: Matern/elementwise -> g (b128 nontemporal stores to HBM + LDS copy).
// Phase 4: in-LDS right-looking Cholesky, column-parallel.
// Phase 5: lower-triangular chol out (b128 nontemporal stores).
__global__ __launch_bounds__(256) void tmap_gram_chol_kernel(
    const float* __restrict__ aug, const float* __restrict__ theta,
    const float* __restrict__ scales, const float* __restrict__ nug_mean,
    const int* __restrict__ batch_idx,
    float* __restrict__ g_out, float* __restrict__ chol_out)
{
  extern __shared__ float smem[];
  float* sXs   = smem;                          // 128 x 50
  float* sS    = smem + N_REPS * XS_STRIDE;     // 128 x 129
  float* sdiag = sS + N_REPS * S_STRIDE;        // 128

  const int n   = blockIdx.x;
  const int tid = threadIdx.x;

  // Per-block scalars (wave-uniform -> compiler hoists to SALU v_s_exp/s_fmac)
  const float et2      = __expf(theta[2]);
  const float len_scal = __expf(theta[5]) * 1.7320508075688772f; // sqrt(2*1.5)
  const float inv_len2 = 1.0f / (len_scal * len_scal);
  const float sig      = __expf(__logf(scales[n]) * theta[4] + theta[3]);
  const float sigma2   = sig * sig;
  const float inv_nug  = 1.0f / nug_mean[n];
  const bool  is_zero  = (batch_idx[n] == 0);

  // ---- Phase 1: gather + NaN-clean + column scale into LDS ----
  for (int e = tid; e < N_REPS * M_NBRS; e += 256) {
    const int r = e / M_NBRS;
    const int m = e - r * M_NBRS;
    float v = aug[((long)r * N_LOCS + n) * (M_NBRS + 1) + 1 + m];
    v = (v != v) ? 0.0f : v;                       // isnan -> 0
    const float scal = __expf(-0.5f * (float)(m + 1) * et2); // sqrt(exp(-k*e^t2))
    sXs[r * XS_STRIDE + m] = v * scal;
  }
  __syncthreads();

  // ---- Phase 2: fp32 WMMA Gram matrix S = Xs * Xs^T ----
  {
    const int wave  = tid >> 5;       // 0..7  -> tile row ti
    const int lane  = tid & 31;
    const int mrow  = lane & 15;      // M (and N) index within tile
    const int khalf = lane >> 4;      // 0: K={0,1}; 1: K={2,3}
    const int ti    = wave;

    // Hoist all 12 A fragments (row-tile ti) — reused across 8 column tiles
    v2f afrag[12];
#pragma unroll
    for (int kk = 0; kk < 12; ++kk) {
      const int k0 = kk * 4 + khalf * 2;
      const float* p = &sXs[(ti * 16 + mrow) * XS_STRIDE + k0];
      afrag[kk].x = p[0];
      afrag[kk].y = p[1];
    }

    for (int tj = 0; tj < 8; ++tj) {
      v8f c = {0.f, 0.f, 0.f, 0.f, 0.f, 0.f, 0.f, 0.f};
#pragma unroll
      for (int kk = 0; kk < 12; ++kk) {
        const int k0 = kk * 4 + khalf * 2;
        // B = (Xs tile tj)^T : B[k][j] = Xs[j][k] -> same LDS addressing as A
        const float* p = &sXs[(tj * 16 + mrow) * XS_STRIDE + k0];
        v2f b;
        b.x = p[0];
        b.y = p[1];
        // (neg_a, A, neg_b, B, c_mod, C, reuse_a, reuse_b)
        c = __builtin_amdgcn_wmma_f32_16x16x4_f32(
                false, afrag[kk], false, b, (short)0, c, false, false);
      }
#pragma unroll
      for (int v = 0; v < 8; ++v) {
        const int row = ti * 16 + v + khalf * 8;   // C layout: VGPR v, lane half
        const int col = tj * 16 + mrow;
        sS[row * S_STRIDE + col] = c[v];
      }
    }
  }
  __syncthreads();

  // ---- Phase 3a: snapshot diagonal (avoids RAW races in 3b) ----
  for (int i = tid; i < N_REPS; i += 256) sdiag[i] = sS[i * S_STRIDE + i];
  __syncthreads();

  // ---- Phase 3b: Matern + assemble g; b128 NT stores + LDS copy ----
  for (int q = tid; q < (N_REPS * N_REPS) / 4; q += 256) {
    const int i  = q >> 5;            // 32 quads per 128-wide row
    const int j0 = (q & 31) << 2;
    const float di = sdiag[i];
    v4f gv;
#pragma unroll
    for (int l = 0; l < 4; ++l) {
      const int j = j0 + l;
      float gval;
      if (is_zero) {
        gval = (i == j) ? 1.0f : 0.0f;
      } else {
        const float sij = sS[i * S_STRIDE + j];
        float sq = (di + sdiag[j] - 2.0f * sij) * inv_len2;
        sq = fmaxf(sq, 0.0f);
        const float sq3d   = 1.7320508075688772f * sqrtf(sq);
        const float matern = (1.0f + sq3d) * __expf(-sq3d);
        gval = (sij + sigma2 * matern) * inv_nug + ((i == j) ? 1.0f : 0.0f);
      }
      gv[l] = gval;
      sS[i * S_STRIDE + j] = gval;   // only this thread touches (i,j); diag via sdiag
    }
    __builtin_nontemporal_store(
        gv, (v4f*)(g_out + (size_t)n * (N_REPS * N_REPS) + i * N_REPS + j0));
  }

  // ---- Phase 4: in-LDS Cholesky (lower), column-parallel right-looking ----
  for (int k = 0; k < N_REPS; ++k) {
    __syncthreads();                       // trailing update of iter k-1 done
    const float dkk = sqrtf(sS[k * S_STRIDE + k]);
    const float inv = 1.0f / dkk;
    __syncthreads();                       // everyone read A[k][k] before write
    for (int i = k + tid; i < N_REPS; i += 256)
      sS[i * S_STRIDE + k] = (i == k) ? dkk : sS[i * S_STRIDE + k] * inv;
    __syncthreads();
    const int j = k + 1 + tid;             // one trailing column per thread
    if (j < N_REPS) {
      const float ljk = sS[j * S_STRIDE + k];
      for (int i = j; i < N_REPS; ++i)
        sS[i * S_STRIDE + j] -= sS[i * S_STRIDE + k] * ljk;
    }
  }
  __syncthreads();

  // ---- Phase 5: emit lower-triangular chol, b128 NT stores ----
  for (int q = tid; q < (N_REPS * N_REPS) / 4; q += 256) {
    const int i  = q >> 5;
    const int j0 = (q & 31) << 2;
    v4f cv;
#pragma unroll
    for (int l = 0; l < 4; ++l) {
      const int j = j0 + l;
      cv[l] = (i >= j) ? sS[i * S_STRIDE + j] : 0.0f;
    }
    __builtin_nontemporal_store(
        cv, (v4f*)(chol_out + (size_t)n * (N_REPS * N_REPS) + i * N_REPS + j0));
  }
}

extern "C" void kernel_launch(void* const* d_in, const int* in_sizes, int n_in,
                              void* d_out, int out_size, void* d_ws, size_t ws_size,
                              hipStream_t stream) {
  const float* aug      = (const float*)d_in[0];
  const float* theta    = (const float*)d_in[1];
  const float* scales   = (const float*)d_in[2];
  const float* nug_mean = (const float*)d_in[3];
  const int*   bidx     = (const int*)d_in[4];

  float* g_out    = (float*)d_out;
  float* chol_out = g_out + (size_t)N_LOCS * N_REPS * N_REPS;

  const size_t lds_bytes =
      (size_t)(N_REPS * XS_STRIDE + N_REPS * S_STRIDE + N_REPS) * sizeof(float); // 92160 B

  tmap_gram_chol_kernel<<<dim3(N_LOCS), dim3(256), lds_bytes, stream>>>(
      aug, theta, scales, nug_mean, bidx, g_out, chol_out);
}